// Model_10350871183718
// MI455X (gfx1250) — compile-verified
//
#include <hip/hip_runtime.h>
#include <hip/hip_bf16.h>
#include <math.h>

// ---------------------------------------------------------------------------
// Rasterizing renderer loss on MI455X (gfx1250).
// Edge functions + depth are affine in pixel coords -> GEMM with K=3, mapped
// onto V_WMMA_F32_16X16X4_F32. Per wave: 2 pixel tiles (32 px), per triangle
// group: 2x global_load_b128 (B-matrix pre-packed in WMMA lane layout by the
// coef kernel) + 8 WMMAs. All tables L2/WGP$-resident (~330KB).
// ---------------------------------------------------------------------------

typedef __attribute__((ext_vector_type(2))) float v2f;
typedef __attribute__((ext_vector_type(8))) float v8f;

#define NF      2560      // triangles
#define NP      16384     // pixels (128x128)
#define NGROUP  160       // NF / 16
#define NBLK    64        // raster blocks: 8 waves x 32 px = 256 px per block
#define BIGF    1.0e10f
#define DISTF   2.732f

// --------------------------- stage 1: coefficients -------------------------
// Per triangle: affine coefficients (A,B,C) for w0,w1,w2 (edge/area) and
// depth. Degenerate triangles folded to "never inside" (w == -1 everywhere).
// Outputs:
//   cf[q*NF + t]  : float4(A,B,C,0)  -- used by the shading stage
//   bl[...]       : B-matrix in WMMA lane layout: for group g, lane l holds
//                   2 float4 = {q0,q1,q2,qd} pairs; lanes 0-15 -> (A,B) of
//                   triangle g*16+l, lanes 16-31 -> (C,0) of g*16+l-16.
__global__ __launch_bounds__(256) void coef_kernel(const float* __restrict__ verts,
                                                   const int* __restrict__ faces,
                                                   float4* __restrict__ cf,
                                                   float4* __restrict__ bl) {
  int t = blockIdx.x * blockDim.x + threadIdx.x;
  if (t >= NF) return;
  int i0 = faces[3 * t + 0], i1 = faces[3 * t + 1], i2 = faces[3 * t + 2];
  // look_at with eye=(2.732,0,0): v' = (v.z, v.y, DIST - v.x)
  float ax = verts[3 * i0 + 2], ay = verts[3 * i0 + 1], az = DISTF - verts[3 * i0 + 0];
  float bx = verts[3 * i1 + 2], by = verts[3 * i1 + 1], bz = DISTF - verts[3 * i1 + 0];
  float cx = verts[3 * i2 + 2], cy = verts[3 * i2 + 1], cz = DISTF - verts[3 * i2 + 0];

  float area = (bx - ax) * (cy - ay) - (by - ay) * (cx - ax);
  bool  valid = fabsf(area) >= 1e-8f;
  float sa = valid ? area : 1e-8f;
  float inv = 1.0f / sa;

  // w0 = edge(b,c,p)/sa ; w1 = edge(c,a,p)/sa ; w2 = edge(a,b,p)/sa
  float A0 = (by - cy) * inv, B0 = (cx - bx) * inv, C0 = ((cy - by) * bx - (cx - bx) * by) * inv;
  float A1 = (cy - ay) * inv, B1 = (ax - cx) * inv, C1 = ((ay - cy) * cx - (ax - cx) * cy) * inv;
  float A2 = (ay - by) * inv, B2 = (bx - ax) * inv, C2 = ((by - ay) * ax - (bx - ax) * ay) * inv;
  if (!valid) {
    A0 = B0 = A1 = B1 = A2 = B2 = 0.0f;
    C0 = C1 = C2 = -1.0f;
  }
  float Ad = A0 * az + A1 * bz + A2 * cz;
  float Bd = B0 * az + B1 * bz + B2 * cz;
  float Cd = C0 * az + C1 * bz + C2 * cz;

  cf[0 * NF + t] = make_float4(A0, B0, C0, 0.0f);
  cf[1 * NF + t] = make_float4(A1, B1, C1, 0.0f);
  cf[2 * NF + t] = make_float4(A2, B2, C2, 0.0f);
  cf[3 * NF + t] = make_float4(Ad, Bd, Cd, 0.0f);

  // B-matrix lane layout: index (g*32 + laneSlot)*2 + half
  int g = t >> 4, c = t & 15;
  bl[(g * 32 + c) * 2 + 0]      = make_float4(A0, B0, A1, B1);   // lanes 0-15: K0,K1
  bl[(g * 32 + c) * 2 + 1]      = make_float4(A2, B2, Ad, Bd);
  bl[(g * 32 + 16 + c) * 2 + 0] = make_float4(C0, 0.0f, C1, 0.0f);  // lanes 16-31: K2,K3
  bl[(g * 32 + 16 + c) * 2 + 1] = make_float4(C2, 0.0f, Cd, 0.0f);
}

// --------------------------- stage 2+3: raster + shade ---------------------
// One wave per 32 pixels (two 16-row WMMA tiles sharing each B-matrix fetch).
__global__ __launch_bounds__(256) void raster_kernel(const float* __restrict__ textures,
                                                     const float* __restrict__ image_ref,
                                                     const float4* __restrict__ cf,
                                                     const float4* __restrict__ bl,
                                                     float* __restrict__ partials) {
  __shared__ float sZ[8][32];
  __shared__ int   sI[8][32];
  __shared__ float sLoss[256];

  const int  lane     = threadIdx.x & 31;
  const int  wave     = threadIdx.x >> 5;
  const int  waveBase = (blockIdx.x * 8 + wave) * 32;  // 32 consecutive pixels
  const bool upper    = lane >= 16;
  const int  col      = lane & 15;

  // A-matrices (16x4 f32): lanes 0-15 hold (K0=px,K1=py) for row M=lane,
  // lanes 16-31 hold (K2=1,K3=0).
  int   p0 = waveBase + col, p1 = waveBase + 16 + col;
  float ax0 = ((float)(p0 & 127) + 0.5f) * (2.0f / 128.0f) - 1.0f;
  float ay0 = 1.0f - ((float)(p0 >> 7) + 0.5f) * (2.0f / 128.0f);
  float ax1 = ((float)(p1 & 127) + 0.5f) * (2.0f / 128.0f) - 1.0f;
  float ay1 = 1.0f - ((float)(p1 >> 7) + 0.5f) * (2.0f / 128.0f);
  v2f AmA, AmB;
  AmA.x = upper ? 1.0f : ax0;  AmA.y = upper ? 0.0f : ay0;
  AmB.x = upper ? 1.0f : ax1;  AmB.y = upper ? 0.0f : ay1;

  float zA[8], zB[8];
  int   iA[8], iB[8];
#pragma unroll
  for (int r = 0; r < 8; ++r) { zA[r] = BIGF; zB[r] = BIGF; iA[r] = 0; iB[r] = 0; }

  const v8f zeroAcc = {};
  const float4* blp = bl + (size_t)lane * 2;  // group stride = 64 float4

  for (int g = 0; g < NGROUP; ++g) {
    if (g + 4 < NGROUP) __builtin_prefetch(&blp[(size_t)(g + 4) * 64], 0, 3);
    float4 h0 = blp[(size_t)g * 64 + 0];
    float4 h1 = blp[(size_t)g * 64 + 1];
    v2f B0, B1, B2, Bd;
    B0.x = h0.x; B0.y = h0.y;   // w0 coefficients
    B1.x = h0.z; B1.y = h0.w;   // w1
    B2.x = h1.x; B2.y = h1.y;   // w2
    Bd.x = h1.z; Bd.y = h1.w;   // depth

    v8f Wa0 = __builtin_amdgcn_wmma_f32_16x16x4_f32(false, AmA, false, B0, (short)0, zeroAcc, false, false);
    v8f Wa1 = __builtin_amdgcn_wmma_f32_16x16x4_f32(false, AmA, false, B1, (short)0, zeroAcc, false, false);
    v8f Wa2 = __builtin_amdgcn_wmma_f32_16x16x4_f32(false, AmA, false, B2, (short)0, zeroAcc, false, false);
    v8f Wad = __builtin_amdgcn_wmma_f32_16x16x4_f32(false, AmA, false, Bd, (short)0, zeroAcc, false, false);
    v8f Wb0 = __builtin_amdgcn_wmma_f32_16x16x4_f32(false, AmB, false, B0, (short)0, zeroAcc, false, false);
    v8f Wb1 = __builtin_amdgcn_wmma_f32_16x16x4_f32(false, AmB, false, B1, (short)0, zeroAcc, false, false);
    v8f Wb2 = __builtin_amdgcn_wmma_f32_16x16x4_f32(false, AmB, false, B2, (short)0, zeroAcc, false, false);
    v8f Wbd = __builtin_amdgcn_wmma_f32_16x16x4_f32(false, AmB, false, Bd, (short)0, zeroAcc, false, false);

    const int t = g * 16 + col;  // this lane's triangle column
#pragma unroll
    for (int r = 0; r < 8; ++r) {
      float mA = fminf(fminf(Wa0[r], Wa1[r]), Wa2[r]);
      float vA = (mA >= 0.0f) ? Wad[r] : BIGF;
      bool  tA = vA < zA[r];                 // strict <: earliest triangle wins
      zA[r] = tA ? vA : zA[r];
      iA[r] = tA ? t  : iA[r];

      float mB = fminf(fminf(Wb0[r], Wb1[r]), Wb2[r]);
      float vB = (mB >= 0.0f) ? Wbd[r] : BIGF;
      bool  tB = vB < zB[r];
      zB[r] = tB ? vB : zB[r];
      iB[r] = tB ? t  : iB[r];
    }
  }

  // Cross-lane argmin over the 16 triangle columns (xor masks stay in-half).
  // Tie-break on smaller index == jnp.argmin first-occurrence semantics.
#pragma unroll
  for (int r = 0; r < 8; ++r) {
    float za = zA[r]; int ia = iA[r];
    float zb = zB[r]; int ib = iB[r];
#pragma unroll
    for (int m = 1; m < 16; m <<= 1) {
      float oza = __shfl_xor(za, m, 32);
      int   oia = __shfl_xor(ia, m, 32);
      bool  ta = (oza < za) || ((oza == za) && (oia < ia));
      za = ta ? oza : za;  ia = ta ? oia : ia;

      float ozb = __shfl_xor(zb, m, 32);
      int   oib = __shfl_xor(ib, m, 32);
      bool  tb = (ozb < zb) || ((ozb == zb) && (oib < ib));
      zb = tb ? ozb : zb;  ib = tb ? oib : ib;
    }
    zA[r] = za; iA[r] = ia;
    zB[r] = zb; iB[r] = ib;
  }
  if (lane == 0) {
#pragma unroll
    for (int r = 0; r < 8; ++r) {
      sZ[wave][r] = zA[r];      sI[wave][r] = iA[r];       // tile0 rows 0-7
      sZ[wave][16 + r] = zB[r]; sI[wave][16 + r] = iB[r];  // tile1 rows 0-7
    }
  }
  if (lane == 16) {
#pragma unroll
    for (int r = 0; r < 8; ++r) {
      sZ[wave][8 + r] = zA[r];  sI[wave][8 + r] = iA[r];   // tile0 rows 8-15
      sZ[wave][24 + r] = zB[r]; sI[wave][24 + r] = iB[r];  // tile1 rows 8-15
    }
  }
  __syncthreads();

  // -------- shading: all 32 lanes, one pixel each -------------------------
  {
    int   p   = waveBase + lane;
    float z   = sZ[wave][lane];
    int   f   = sI[wave][lane];
    bool  hit = z < (BIGF * 0.5f);
    float px  = ((float)(p & 127) + 0.5f) * (2.0f / 128.0f) - 1.0f;
    float py  = 1.0f - ((float)(p >> 7) + 0.5f) * (2.0f / 128.0f);

    float4 q0 = cf[0 * NF + f];
    float4 q1 = cf[1 * NF + f];
    float4 q2 = cf[2 * NF + f];
    float u0 = fmaf(q0.x, px, fmaf(q0.y, py, q0.z));
    float u1 = fmaf(q1.x, px, fmaf(q1.y, py, q1.z));
    float u2 = fmaf(q2.x, px, fmaf(q2.y, py, q2.z));
    u0 = fminf(fmaxf(u0, 0.0f), 1.0f);
    u1 = fminf(fmaxf(u1, 0.0f), 1.0f);
    u2 = fminf(fmaxf(u2, 0.0f), 1.0f);
    float inv  = 1.0f / (u0 + u1 + u2 + 1e-8f);
    float pos0 = u0 * inv * 3.0f, pos1 = u1 * inv * 3.0f, pos2 = u2 * inv * 3.0f;
    int l0 = (int)floorf(pos0); l0 = l0 < 0 ? 0 : (l0 > 2 ? 2 : l0);
    int l1 = (int)floorf(pos1); l1 = l1 < 0 ? 0 : (l1 > 2 ? 2 : l1);
    int l2 = (int)floorf(pos2); l2 = l2 < 0 ? 0 : (l2 > 2 ? 2 : l2);
    float f0 = pos0 - (float)l0, f1 = pos1 - (float)l1, f2 = pos2 - (float)l2;

    const float* tb = textures + (size_t)f * 192;  // 4*4*4*3 floats per face
    float c0 = 0.0f, c1 = 0.0f, c2 = 0.0f;
#pragma unroll
    for (int di = 0; di < 2; ++di)
#pragma unroll
      for (int dj = 0; dj < 2; ++dj)
#pragma unroll
        for (int dk = 0; dk < 2; ++dk) {
          float wg = (di ? f0 : 1.0f - f0) * (dj ? f1 : 1.0f - f1) * (dk ? f2 : 1.0f - f2);
          const float* cp = tb + (size_t)(((l0 + di) * 16 + (l1 + dj) * 4 + (l2 + dk)) * 3);
          c0 += wg * tanhf(cp[0]);
          c1 += wg * tanhf(cp[1]);
          c2 += wg * tanhf(cp[2]);
        }
    c0 = hit ? c0 : 0.0f;
    c1 = hit ? c1 : 0.0f;
    c2 = hit ? c2 : 0.0f;
    float d0 = c0 - image_ref[0 * NP + p];
    float d1 = c1 - image_ref[1 * NP + p];
    float d2 = c2 - image_ref[2 * NP + p];
    sLoss[threadIdx.x] = d0 * d0 + d1 * d1 + d2 * d2;
  }
  __syncthreads();

  // Deterministic block tree-reduction.
#pragma unroll
  for (int s = 128; s > 0; s >>= 1) {
    if ((int)threadIdx.x < s) sLoss[threadIdx.x] += sLoss[threadIdx.x + s];
    __syncthreads();
  }
  if (threadIdx.x == 0) partials[blockIdx.x] = sLoss[0];
}

// --------------------------- stage 4: finalize -----------------------------
__global__ void finalize_kernel(const float* __restrict__ partials, float* __restrict__ out) {
  if (threadIdx.x == 0 && blockIdx.x == 0) {
    float s = 0.0f;
    for (int i = 0; i < NBLK; ++i) s += partials[i];  // fixed order: deterministic
    out[0] = s;
  }
}

extern "C" void kernel_launch(void* const* d_in, const int* in_sizes, int n_in,
                              void* d_out, int out_size, void* d_ws, size_t ws_size,
                              hipStream_t stream) {
  const float* verts     = (const float*)d_in[0];  // (1,1320,3)
  const float* textures  = (const float*)d_in[1];  // (1,2560,4,4,4,3)
  const float* image_ref = (const float*)d_in[2];  // (1,3,128,128)
  const int*   faces     = (const int*)d_in[3];    // (1,2560,3)
  float*       out       = (float*)d_out;          // scalar loss

  float4* cf       = (float4*)d_ws;                                        // 4*NF float4
  float4* bl       = (float4*)((char*)d_ws + (size_t)4 * NF * 16);         // NGROUP*32*2 float4
  float*  partials = (float*)((char*)d_ws + (size_t)8 * NF * 16);          // NBLK floats

  coef_kernel<<<(NF + 255) / 256, 256, 0, stream>>>(verts, faces, cf, bl);
  raster_kernel<<<NBLK, 256, 0, stream>>>(textures, image_ref, cf, bl, partials);
  finalize_kernel<<<1, 32, 0, stream>>>(partials, out);
}